// HyperPatchNoPadding_29961691857147
// MI455X (gfx1250) — compile-verified
//
#include <hip/hip_runtime.h>

typedef float v2f __attribute__((ext_vector_type(2)));
typedef float v8f __attribute__((ext_vector_type(8)));

#define NC_   64
#define NH_   256
#define NW_   256
#define NOUT_ 64
#define NFH_  8
#define NFW_  8
#define PH_   32
#define PW_   32
#define PLANE (NH_ * NW_)   /* 65536 elements per channel image */
#define CPR   160           /* LDS dwords per channel-pair row: 128 data + 32 pad; 160%64==32 */
#define TEAMSZ (32 * CPR)   /* dwords per (buf, team) X tile: 32 channel-pairs */

// One workgroup per (batch, fh, fw) cell: GEMM  Y[64x1024] = W[64x64] * X[64x1024]
// 8 waves, 2 teams of 4: team = N-half (512 pixels), wave-in-team = 16-row M-tile.
// X staged to LDS channel-pair-interleaved via double-buffered async b32 copies (GVS form:
// per-lane pixel offset in VGPR, per-issue channel/row displacement in SGPR base), so each
// WMMA B-fragment is one aligned, bank-conflict-free ds_load_b64 (fused to ds_load_2addr_b64).
__global__ __launch_bounds__(256, 2) void hyperpatch_wmma_kernel(
    const float* __restrict__ x, const float* __restrict__ wt, float* __restrict__ y)
{
    __shared__ float ldsW[NOUT_ * NC_];                   // 16 KB  W[o][i]
    __shared__ __align__(16) float ldsX[2][2][TEAMSZ];    // 80 KB  [buf][team][cpair][pix*2+par]

    const int cell = blockIdx.x;          // 0..511
    const int b = cell >> 6;
    const int f = (cell >> 3) & 7;
    const int g = cell & 7;
    const int tid  = threadIdx.x;
    const int wave = tid >> 5;
    const int lane = tid & 31;
    const int mi    = wave & 3;           // M-tile index within team
    const int m0    = mi * 16;
    const int nhalf = wave >> 2;          // team: which 512-pixel half of the patch
    const int lq    = lane & 15;          // column within a 16-wide N tile
    const int chH   = lane >> 4;          // K sub-select: lanes 0-15 -> K0/K1, 16-31 -> K2/K3
    const int koff  = chH * 2;

    const size_t base = (size_t)b * NC_ * PLANE;          // same for x (C=64) and y (OUT=64)
    const float* xb = x + base;
    float*       yb = y + base;

    const unsigned ldsx0 = (unsigned)(uintptr_t)&ldsX[0][0][0];  // LDS byte offset (addr[31:0])
    const unsigned vpix  = (unsigned)(lane * 4);          // per-lane global byte offset (column)

    // team-cooperative async staging: 32 x b32 per wave, one channel-row (32 px) per issue.
    // dest dword = (c>>1)*CPR + 2*pix + (c&1)  (pix = r*32 + lane); wave mi owns ch mi*16..+15.
    auto issue_block = [&](int nb, int buf) {
        const int p0 = nhalf * 16 + nb * 2;
        // wave-uniform global base: (row p0, channel mi*16), forced into SGPRs
        const unsigned long long gb = (unsigned long long)(uintptr_t)
            (xb + (size_t)(mi * 16) * PLANE + (f * PH_ + p0) * NW_ + g * PW_);
        const unsigned long long gbase =
            ((unsigned long long)__builtin_amdgcn_readfirstlane((unsigned)(gb >> 32)) << 32) |
             (unsigned long long)__builtin_amdgcn_readfirstlane((unsigned)gb);
        const unsigned lbase = ldsx0 + (unsigned)((buf * 2 + nhalf) * TEAMSZ * 4)
                             + (unsigned)(mi * 8 * CPR * 4) + (unsigned)(lane * 8);
        #pragma unroll
        for (int i = 0; i < 32; ++i) {
            const int cl = i >> 1;                         // channel - mi*16 (uniform per issue)
            const int r  = i & 1;                          // row within 2-row pixel block
            const unsigned la = lbase +
                (unsigned)((((cl >> 1) * CPR) + r * 64 + (cl & 1)) * 4);
            const unsigned long long ga = gbase
                + (unsigned long long)cl * (PLANE * 4) + (unsigned)(r * NW_ * 4);
            asm volatile("global_load_async_to_lds_b32 %0, %1, %2"
                         :: "v"(la), "v"(vpix), "s"(ga) : "memory");
        }
    };

    issue_block(0, 0);                    // start DMA of first block immediately

    // ---- stage weights: weight[b][o*C+i][f][g], stride FH*FW=64 floats (L2-resident gather)
    {
        const float* wp = wt + (size_t)b * (NOUT_ * NC_) * (NFH_ * NFW_) + f * NFW_ + g;
        #pragma unroll
        for (int k = 0; k < 16; ++k) {
            int m = tid + k * 256;                         // m = o*64 + i
            ldsW[m] = wp[(size_t)m * (NFH_ * NFW_)];
        }
    }
    __syncthreads();

    // ---- hoist all A fragments (16 k-steps): rows m0..m0+15, K=4s..4s+3
    v2f afrag[16];
    {
        const float* wr = &ldsW[(m0 + lq) * NC_ + koff];
        #pragma unroll
        for (int s = 0; s < 16; ++s)
            afrag[s] = *(const v2f*)(wr + 4 * s);          // 8B-aligned LDS read
    }

    for (int nb = 0; nb < 8; ++nb) {
        const int cur = nb & 1;
        if (nb < 7) {
            issue_block(nb + 1, 1 - cur);                  // DMA next block into other buffer
            asm volatile("s_wait_asynccnt 0x20" ::: "memory");  // this block's 32 copies done
        } else {
            asm volatile("s_wait_asynccnt 0x0" ::: "memory");   // drain last block
        }
        __syncthreads();                                   // team-wide visibility of buf[cur]

        // per-lane fragment base: chpair chH, pixel column lq
        const float* xt = &ldsX[cur][nhalf][0] + chH * CPR + 2 * lq;
        const int p0 = nhalf * 16 + nb * 2;

        v8f acc[4] = {v8f{}, v8f{}, v8f{}, v8f{}};
        #pragma unroll
        for (int s = 0; s < 16; ++s) {                     // K = 4s .. 4s+3
            v2f bf[4];
            #pragma unroll
            for (int t = 0; t < 4; ++t)                    // one ds_load_b64 per fragment
                bf[t] = *(const v2f*)(xt + 2 * s * CPR + t * 32);
            #pragma unroll
            for (int t = 0; t < 4; ++t) {
                acc[t] = __builtin_amdgcn_wmma_f32_16x16x4_f32(
                    /*neg_a=*/false, afrag[s],
                    /*neg_b=*/false, bf[t],
                    /*c_mod=*/(short)0, acc[t],
                    /*reuse_a=*/false, /*reuse_b=*/false);
            }
        }

        // ---- store: C/D layout: VGPR v -> row m0+v (lanes 0-15) / m0+v+8 (lanes 16-31)
        const int orow = m0 + (koff << 2);
        #pragma unroll
        for (int t = 0; t < 4; ++t) {
            const int pix = (f * PH_ + p0 + (t >> 1)) * NW_ + g * PW_ + (t & 1) * 16 + lq;
            float* yp = yb + pix;
            #pragma unroll
            for (int v = 0; v < 8; ++v)
                yp[(size_t)(orow + v) * PLANE] = acc[t][v];
        }
        __syncthreads();                                   // buf[cur] free before re-issue
    }
}

extern "C" void kernel_launch(void* const* d_in, const int* in_sizes, int n_in,
                              void* d_out, int out_size, void* d_ws, size_t ws_size,
                              hipStream_t stream) {
    const float* x  = (const float*)d_in[0];   // [8, 64, 256, 256] f32
    const float* wt = (const float*)d_in[1];   // [8, 4096, 8, 8]   f32
    float* y = (float*)d_out;                  // [8, 64, 256, 256] f32
    (void)in_sizes; (void)n_in; (void)out_size; (void)d_ws; (void)ws_size;

    hyperpatch_wmma_kernel<<<dim3(8 * NFH_ * NFW_), dim3(256), 0, stream>>>(x, wt, y);
}